// GBLoss_8942121910839
// MI455X (gfx1250) — compile-verified
//
#include <hip/hip_runtime.h>

// Problem constants (match the reference exactly).
#define ROWS 8192
#define COLS 4096
#define TPB  256
#define EPT  16          // elements per thread = COLS / TPB
#define NWAVE (TPB / 32) // wave32 on gfx1250
#define TOPK 15
#define NEG_INF (-__builtin_huge_valf())

typedef unsigned int uint32x4 __attribute__((ext_vector_type(4)));
typedef int          int32x4  __attribute__((ext_vector_type(4)));
typedef int          int32x8  __attribute__((ext_vector_type(8)));

// One TDM descriptor: contiguous 1-row tile (COLS f32) global -> LDS.
// Layout per CDNA5 ISA sec 8.3/8.4:
//   group0: [1:0]=count=1, [63:32]=lds_addr, [120:64]=global_addr, [127:126]=type=2
//   group1: [17:16]=data_size(2 -> 4B), [79:48]=tensor_dim0, [111:80]=tensor_dim1,
//           [127:112]=tile_dim0, [143:128]=tile_dim1, [207:160]=tensor_dim0_stride
__device__ __forceinline__ void tdm_load_row(const float* gsrc, unsigned lds_off)
{
    unsigned long long ga = (unsigned long long)gsrc;
    uint32x4 g0;
    g0[0] = 1u;                                              // count=1 (user descriptor)
    g0[1] = lds_off;                                         // LDS byte address
    g0[2] = (unsigned)(ga & 0xFFFFFFFFu);                    // global_addr[31:0]
    g0[3] = (unsigned)((ga >> 32) & 0x01FFFFFFu) | (2u << 30); // addr[56:32] | type=2

    int32x8 g1;
    g1[0] = (2 << 16);        // workgroup_mask=0, data_size=2 (4 bytes)
    g1[1] = (COLS << 16);     // tensor_dim0[15:0] in bits[31:16]
    g1[2] = (1 << 16);        // tensor_dim0[31:16]=0 ; tensor_dim1[15:0]=1
    g1[3] = (COLS << 16);     // tensor_dim1[31:16]=0 ; tile_dim0=COLS
    g1[4] = 1;                // tile_dim1=1, tile_dim2=0
    g1[5] = COLS;             // tensor_dim0_stride[31:0]
    g1[6] = 0;                // stride0[47:32]=0, tensor_dim1_stride[15:0]=0
    g1[7] = 0;

    int32x4 z4 = {0, 0, 0, 0};
#if __clang_major__ >= 23
    int32x8 z8 = {0, 0, 0, 0, 0, 0, 0, 0};
    __builtin_amdgcn_tensor_load_to_lds(g0, g1, z4, z4, z8, 0);
#else
    __builtin_amdgcn_tensor_load_to_lds(g0, g1, z4, z4, 0);
#endif
}

// One block per TWO rows, software-pipelined via the Tensor Data Mover:
// wave 0 issues both row DMAs, row p's compute overlaps row p+1's DMA.
__global__ __launch_bounds__(TPB) void topk_ce_rows(const float* __restrict__ x,
                                                    const int* __restrict__ y,
                                                    float* __restrict__ row_loss)
{
    __shared__ __align__(16) float s_row[2][COLS];   // 32 KiB staging (TDM dest)
    __shared__ float s_wmax[2][NWAVE];               // double-buffered per-wave maxima
    __shared__ int   s_claim[2][TOPK + 1];           // one claim slot per round per row
    __shared__ float s_xgt[2];

    const int tid  = threadIdx.x;
    const int lane = tid & 31;
    const int wid  = tid >> 5;
    const int row0 = blockIdx.x * 2;

    if (tid <= TOPK) { s_claim[0][tid] = 0; s_claim[1][tid] = 0; }

    // Wave 0 only: TDM ignores EXEC, so the issue must be branch-confined to
    // one wave or the DMA would be duplicated 8x.
    if (wid == 0) {
        const unsigned lbase = (unsigned)(unsigned long long)&s_row[0][0];
        tdm_load_row(x + (size_t)row0 * COLS,       lbase);
        tdm_load_row(x + (size_t)(row0 + 1) * COLS, lbase + COLS * sizeof(float));
    }

    #pragma unroll 1
    for (int p = 0; p < 2; ++p) {
        const int row = row0 + p;
        const int c   = y[row];

        // TENSORcnt decrements in-order: <=1 means row0 landed, ==0 means row1 landed.
        if (wid == 0) {
            if (p == 0) __builtin_amdgcn_s_wait_tensorcnt(1);
            else        __builtin_amdgcn_s_wait_tensorcnt(0);
        }
        __syncthreads();   // row p now visible to all waves (also orders claim init)

        // 128-bit LDS reads; capture ground-truth logit and mask it to -inf.
        const float4* rowq = (const float4*)&s_row[p][0];
        float v[EPT];
        #pragma unroll
        for (int j = 0; j < EPT / 4; ++j) {
            float4 q = rowq[tid + TPB * j];
            const int base = 4 * (tid + TPB * j);
            const int d = c - base;
            if (d >= 0 && d < 4) {
                float g = (d == 0) ? q.x : (d == 1) ? q.y : (d == 2) ? q.z : q.w;
                s_xgt[p] = g;
                if      (d == 0) q.x = NEG_INF;
                else if (d == 1) q.y = NEG_INF;
                else if (d == 2) q.z = NEG_INF;
                else             q.w = NEG_INF;
            }
            v[4 * j + 0] = q.x; v[4 * j + 1] = q.y;
            v[4 * j + 2] = q.z; v[4 * j + 3] = q.w;
        }

        // Cached per-thread (max, index) over the 16 register-resident values.
        float lm = v[0];
        int   li = 0;
        #pragma unroll
        for (int i = 1; i < EPT; ++i)
            if (v[i] > lm) { lm = v[i]; li = i; }

        unsigned live = 0xFFFFu;
        float M = 0.f, s = 0.f, xgt = 0.f;

        #pragma unroll 1
        for (int r = 0; r < TOPK; ++r) {
            // wave32 butterfly max of the cached candidates
            float wm = lm;
            #pragma unroll
            for (int off = 16; off > 0; off >>= 1)
                wm = fmaxf(wm, __shfl_xor(wm, off, 32));
            if (lane == 0) s_wmax[r & 1][wid] = wm;
            __syncthreads();

            // uniform block max from the 8 per-wave maxima
            float bmax = s_wmax[r & 1][0];
            #pragma unroll
            for (int w = 1; w < NWAVE; ++w)
                bmax = fmaxf(bmax, s_wmax[r & 1][w]);

            // fold logsumexp into the rounds; M is known after round 0
            if (r == 0) {
                xgt = s_xgt[p];
                M   = fmaxf(xgt, bmax);
                s   = expf(xgt - M) + expf(bmax - M);
            } else {
                s  += expf(bmax - M);
            }

            // elect exactly one claimer block-wide; only it rescans its registers
            if (lm == bmax) {
                if (atomicAdd(&s_claim[p][r], 1) == 0) {
                    live &= ~(1u << li);
                    float nl = NEG_INF;
                    int   ni = 0;
                    #pragma unroll
                    for (int i = 0; i < EPT; ++i) {
                        const float cv = ((live >> i) & 1u) ? v[i] : NEG_INF;
                        if (cv > nl) { nl = cv; ni = i; }
                    }
                    lm = nl; li = ni;
                }
            }
            // double-buffered s_wmax + one barrier/round keeps writes of round
            // r+1 safe against stragglers still reading round r's buffer
        }

        if (tid == 0) row_loss[row] = (M + logf(s)) - xgt;
    }
}

// Deterministic mean over the per-row losses (fixed summation order).
__global__ __launch_bounds__(TPB) void mean_reduce(const float* __restrict__ in,
                                                   float* __restrict__ out, int n)
{
    __shared__ float sh[NWAVE];
    const int tid = threadIdx.x;
    float acc = 0.f;
    for (int i = tid; i < n; i += TPB) acc += in[i];
    #pragma unroll
    for (int off = 16; off > 0; off >>= 1) acc += __shfl_xor(acc, off, 32);
    if ((tid & 31) == 0) sh[tid >> 5] = acc;
    __syncthreads();
    if (tid == 0) {
        float t = 0.f;
        #pragma unroll
        for (int w = 0; w < NWAVE; ++w) t += sh[w];
        out[0] = t / (float)n;
    }
}

extern "C" void kernel_launch(void* const* d_in, const int* in_sizes, int n_in,
                              void* d_out, int out_size, void* d_ws, size_t ws_size,
                              hipStream_t stream)
{
    (void)n_in; (void)out_size; (void)ws_size;
    const float* x = (const float*)d_in[0];   // [8192, 4096] f32
    const int*   y = (const int*)d_in[1];     // [8192] i32 labels
    float* losses  = (float*)d_ws;            // 8192 floats of scratch
    float* out     = (float*)d_out;           // scalar f32

    const int B = in_sizes[1];                // 8192 rows, 2 rows per block

    topk_ce_rows<<<B / 2, TPB, 0, stream>>>(x, y, losses);
    mean_reduce<<<1, TPB, 0, stream>>>(losses, out, B);
}